// MBFE_55542517072420
// MI455X (gfx1250) — compile-verified
//
#include <hip/hip_runtime.h>
#include <math.h>
#include <stdint.h>

#define SDIM 1024
#define SS (SDIM*SDIM)
#define NCH 8
#define BWD 170            // 1024 // 6
#define PIF 3.14159265358979323846f

struct alignas(8) cpx { float x, y; };

typedef __attribute__((ext_vector_type(2))) float v2f;
typedef __attribute__((ext_vector_type(8))) float v8f;

__device__ __forceinline__ cpx cmk(float a, float b){ cpx r; r.x=a; r.y=b; return r; }
__device__ __forceinline__ cpx cadd(cpx a, cpx b){ return cmk(a.x+b.x, a.y+b.y); }
__device__ __forceinline__ cpx csub(cpx a, cpx b){ return cmk(a.x-b.x, a.y-b.y); }
__device__ __forceinline__ cpx cmul(cpx a, cpx b){ return cmk(a.x*b.x - a.y*b.y, a.x*b.y + a.y*b.x); }
__device__ __forceinline__ cpx cnegi(cpx a){ return cmk(a.y, -a.x); } // -i * a

// ---- CDNA5 async global<->LDS plumbing -------------------------------------
#if defined(__has_builtin)
# if __has_builtin(__builtin_amdgcn_global_load_async_to_lds_b128) && \
     __has_builtin(__builtin_amdgcn_global_store_async_from_lds_b128)
#  define HAVE_ASYNC_LDS 1
# endif
#endif

// builtin expects: (vector_size(16) int __device__ *global, addrspace(3) counterpart, imm offset, cpol)
typedef __attribute__((__vector_size__(4 * sizeof(int)))) int i32x4;
typedef __attribute__((address_space(1))) i32x4 gi32x4;
typedef __attribute__((address_space(3))) i32x4 li32x4;

__device__ __forceinline__ gi32x4* as_global(void* p) {
  return (gi32x4*)(unsigned long long)(uintptr_t)p;
}
__device__ __forceinline__ li32x4* as_lds(void* p) {
  // flat LDS aperture: addr[31:0] is the in-allocation byte offset
  return (li32x4*)(unsigned)(uintptr_t)p;
}
__device__ __forceinline__ void wait_async0() {
#if defined(__has_builtin) && __has_builtin(__builtin_amdgcn_s_wait_asynccnt)
  __builtin_amdgcn_s_wait_asynccnt(0);
#else
  asm volatile("s_wait_asynccnt 0x0" ::: "memory");
#endif
}

// ---------------------------------------------------------------------------
// 1) Edge blur: alpha*img + (1-alpha)*gauss7x7(img), zero-padded conv.
//    Writes result as complex (imag=0) so the FFT reads it directly.
// ---------------------------------------------------------------------------
__global__ __launch_bounds__(256) void blur_kernel(const float* __restrict__ lms,
                                                   const float* __restrict__ fuse,
                                                   cpx* __restrict__ outL,
                                                   cpx* __restrict__ outF) {
  int idx = blockIdx.x * blockDim.x + threadIdx.x;
  if (idx >= SS) return;
  int r = idx >> 10, c = idx & (SDIM - 1);

  float w[7]; float wsum = 0.f;
  #pragma unroll
  for (int i = 0; i < 7; i++) { float d = (float)i - 3.f; w[i] = __expf(-0.5f * d * d); wsum += w[i]; }
  float winv = 1.f / wsum;
  #pragma unroll
  for (int i = 0; i < 7; i++) w[i] *= winv;

  float bl = 0.f, bf = 0.f;
  #pragma unroll
  for (int i = 0; i < 7; i++) {
    int rr = r - 3 + i;
    if (rr < 0 || rr >= SDIM) continue;
    #pragma unroll
    for (int j = 0; j < 7; j++) {
      int cc = c - 3 + j;
      if (cc < 0 || cc >= SDIM) continue;
      float wij = w[i] * w[j];
      bl += wij * lms[rr * SDIM + cc];
      bf += wij * fuse[rr * SDIM + cc];
    }
  }

  int gr = (r < BWD) ? r : ((SDIM - 1 - r) < BWD ? (SDIM - 1 - r) : -1);
  int gc = (c < BWD) ? c : ((SDIM - 1 - c) < BWD ? (SDIM - 1 - c) : -1);
  int m = gr > gc ? gr : gc;
  float a = (m >= 0) ? (float)m * (1.f / (float)BWD) : 1.f;

  outL[idx] = cmk(a * lms[idx]  + (1.f - a) * bl, 0.f);
  outF[idx] = cmk(a * fuse[idx] + (1.f - a) * bf, 0.f);
}

// ---------------------------------------------------------------------------
// 2) 1024-point row FFT. 1024 = 64 x 16 four-step:
//    stage A: 16 x (64-pt Stockham radix-4, 3 stages) over n2
//    stage B: twiddle W1024^(n1*k2), restore k2-major layout
//    stage C: 16-pt DFT over n1 as complex matmul via v_wmma_f32_16x16x4_f32
//             all 8 waves active: wave = (tile<<1)|part, part 0=Re / 1=Im.
//    Line moves global<->LDS with 128-bit ASYNCcnt transfers when available.
// ---------------------------------------------------------------------------
__global__ __launch_bounds__(256) void fft_row_kernel(cpx* __restrict__ A,
                                                      cpx* __restrict__ B,
                                                      int rowsA) {
  __shared__ float reP[SDIM], imP[SDIM], reQ[SDIM], imQ[SDIM];
  __shared__ cpx   stage[SDIM];          // interleaved staging for async DMA

  int row = blockIdx.x;
  cpx* line = (row < rowsA) ? (A + (size_t)row * SDIM)
                            : (B + (size_t)(row - rowsA) * SDIM);
  int t = threadIdx.x;

#ifdef HAVE_ASYNC_LDS
  #pragma unroll
  for (int v = 0; v < 2; v++) {
    int e = (t + 256 * v) * 2;           // 2 cpx = 16B per async b128
    __builtin_amdgcn_global_load_async_to_lds_b128(as_global(line + e),
                                                   as_lds(&stage[e]), 0, 0);
  }
  wait_async0();
  __syncthreads();
  #pragma unroll
  for (int v = 0; v < 4; v++) {
    int i = t + 256 * v;
    reP[i] = stage[i].x; imP[i] = stage[i].y;
  }
#else
  #pragma unroll
  for (int v = 0; v < 4; v++) {
    int i = t + 256 * v;
    cpx z = line[i];
    reP[i] = z.x; imP[i] = z.y;
  }
#endif
  __syncthreads();

  // ---- stage A: 16 sub-sequences (n1 = s), each length-64 over n2.
  // element e of sub-seq s lives at LDS index s + 16*e.
  int s = t & 15, j = t >> 4;     // j in [0,16): one radix-4 butterfly per stage
  float *sr = reP, *si = imP, *dr = reQ, *di = imQ;
  #pragma unroll
  for (int L = 1; L <= 16; L *= 4) {
    int k = j & (L - 1);
    int base = s + 16 * j;                    // e_in = j + 16*q  -> idx base + 256*q
    cpx a0 = cmk(sr[base      ], si[base      ]);
    cpx a1 = cmk(sr[base + 256], si[base + 256]);
    cpx a2 = cmk(sr[base + 512], si[base + 512]);
    cpx a3 = cmk(sr[base + 768], si[base + 768]);
    float sn, cs;
    __sincosf(-PIF * (float)k / (2.f * (float)L), &sn, &cs);
    cpx w1 = cmk(cs, sn), w2 = cmul(w1, w1), w3 = cmul(w2, w1);
    a1 = cmul(a1, w1); a2 = cmul(a2, w2); a3 = cmul(a3, w3);
    cpx p02 = cadd(a0, a2), m02 = csub(a0, a2);
    cpx p13 = cadd(a1, a3), m13 = csub(a1, a3);
    cpx jm  = cnegi(m13);
    int eo = k + 4 * (j / L) * L;             // output e = eo + q*L
    int ob = s + 16 * eo;
    int st = 16 * L;
    cpx y0 = cadd(p02, p13); dr[ob       ] = y0.x; di[ob       ] = y0.y;
    cpx y1 = cadd(m02, jm ); dr[ob +   st] = y1.x; di[ob +   st] = y1.y;
    cpx y2 = csub(p02, p13); dr[ob + 2*st] = y2.x; di[ob + 2*st] = y2.y;
    cpx y3 = csub(m02, jm ); dr[ob + 3*st] = y3.x; di[ob + 3*st] = y3.y;
    __syncthreads();
    float* tp;
    tp = sr; sr = dr; dr = tp;
    tp = si; si = di; di = tp;
  }
  // after 3 stages Y[n1][k2] is in sr/si at index n1 + 16*k2; dr/di are free.

  // ---- stage B: twiddle W1024^(n1*k2); re-store k2-major: idx k2 + 64*n1
  #pragma unroll
  for (int v = 0; v < 4; v++) {
    int i = t + 256 * v;
    int n1 = i & 15, k2 = i >> 4;
    cpx y = cmk(sr[i], si[i]);
    int p = (n1 * k2) & 1023;
    float sn, cs;
    __sincosf(-(2.f * PIF / 1024.f) * (float)p, &sn, &cs);
    y = cmul(y, cmk(cs, sn));
    dr[k2 + 64 * n1] = y.x;
    di[k2 + 64 * n1] = y.y;
  }
  __syncthreads();

  // ---- stage C: Z[k1][k2] = sum_n1 DFT16[k1][n1] * Y[n1][k2]
  // wave w: k2-tile = w>>1, part = w&1 (0 -> Re(Z), 1 -> Im(Z)).
  // Re(Z) = cos*Yr + sin*Yi ; Im(Z) = cos*Yi - sin*Yr   (DFT16 = cos - i sin)
  {
    int wave = t >> 5, lane = t & 31;
    int tile = wave >> 1;
    int part = wave & 1;
    int lh = lane & 15;
    int hi = lane >> 4;                 // 0 or 1
    int col = tile * 16 + lh;           // k2
    float* bx = part ? di : dr;         // multiplied by cos
    float* by = part ? dr : di;         // multiplied by +/- sin
    float sgn = part ? -1.f : 1.f;
    v8f acc = {0.f,0.f,0.f,0.f,0.f,0.f,0.f,0.f};
#if defined(__has_builtin) && __has_builtin(__builtin_amdgcn_wmma_f32_16x16x4_f32)
    #pragma unroll
    for (int kc = 0; kc < 4; kc++) {
      int kk = kc * 4 + 2 * hi;         // ISA 16x4 A layout: VGPR0 K={0,2}, VGPR1 K={1,3}
      int mk0 = (lh * kk) & 15;
      int mk1 = (lh * (kk + 1)) & 15;
      float s0, c0, s1, c1;
      __sincosf((PIF / 8.f) * (float)mk0, &s0, &c0);
      __sincosf((PIF / 8.f) * (float)mk1, &s1, &c1);
      v2f ac = {c0, c1};
      v2f as = {sgn * s0, sgn * s1};
      v2f b0 = {bx[col + 64 * kk], bx[col + 64 * (kk + 1)]};
      v2f b1 = {by[col + 64 * kk], by[col + 64 * (kk + 1)]};
      acc = __builtin_amdgcn_wmma_f32_16x16x4_f32(false, ac, false, b0, (short)0, acc, false, false);
      acc = __builtin_amdgcn_wmma_f32_16x16x4_f32(false, as, false, b1, (short)0, acc, false, false);
    }
#else
    // scalar fallback: direct 16-point DFT (this lane's D slice, one part)
    for (int rr = 0; rr < 8; rr++) {
      int k1 = rr + 8 * hi;
      float a = 0.f;
      for (int n1 = 0; n1 < 16; n1++) {
        float sn, cs;
        __sincosf((PIF / 8.f) * (float)((k1 * n1) & 15), &sn, &cs);
        a += cs * bx[col + 64 * n1] + sgn * sn * by[col + 64 * n1];
      }
      acc[rr] = a;
    }
#endif
    // D layout: VGPR rr holds M = rr + 8*hi, N = lh. Final index k = k2 + 64*k1.
    float* outp = part ? si : sr;
    #pragma unroll
    for (int rr = 0; rr < 8; rr++) {
      int k1 = rr + 8 * hi;
      outp[col + 64 * k1] = acc[rr];
    }
  }
  __syncthreads();

#ifdef HAVE_ASYNC_LDS
  #pragma unroll
  for (int v = 0; v < 4; v++) {
    int i = t + 256 * v;
    stage[i] = cmk(sr[i], si[i]);
  }
  __syncthreads();
  #pragma unroll
  for (int v = 0; v < 2; v++) {
    int e = (t + 256 * v) * 2;
    __builtin_amdgcn_global_store_async_from_lds_b128(as_global(line + e),
                                                      as_lds(&stage[e]), 0, 0);
  }
  wait_async0();
#else
  #pragma unroll
  for (int v = 0; v < 4; v++) {
    int i = t + 256 * v;
    line[i] = cmk(sr[i], si[i]);
  }
#endif
}

// ---------------------------------------------------------------------------
// 3) Tiled complex transpose (float2, 32x32 tile, padded LDS).
// ---------------------------------------------------------------------------
__global__ __launch_bounds__(256) void transpose_kernel(const cpx* __restrict__ src,
                                                        cpx* __restrict__ dst) {
  __shared__ cpx tile[32][33];
  int bx = blockIdx.x * 32, by = blockIdx.y * 32;
  int tx = threadIdx.x, ty = threadIdx.y;
  #pragma unroll
  for (int r = ty; r < 32; r += 8)
    tile[r][tx] = src[(size_t)(by + r) * SDIM + bx + tx];
  __syncthreads();
  #pragma unroll
  for (int r = ty; r < 32; r += 8)
    dst[(size_t)(bx + r) * SDIM + by + tx] = tile[tx][r];
}

// ---------------------------------------------------------------------------
// 4) Wiener quotient (in transposed spectrum space; grad is transpose-symmetric).
//    Writes conj(Q) so the inverse FFT can reuse the forward kernel.
// ---------------------------------------------------------------------------
__global__ __launch_bounds__(256) void wiener_kernel(const cpx* __restrict__ FL,
                                                     const cpx* __restrict__ FF,
                                                     cpx* __restrict__ Qc) {
  int idx = blockIdx.x * blockDim.x + threadIdx.x;
  if (idx >= SS) return;
  int i = idx >> 10, jv = idx & (SDIM - 1);
  float u = (float)i, v = (float)jv;
  // |fft(gv)|^2 + |fft(gh)|^2, gv = [-1,0,1]^T (x) [1,2,1]
  float Au = 2.f - 2.f * __cosf((4.f * PIF / 1024.f) * u);
  float Bv = 2.f + 2.f * __cosf((2.f * PIF / 1024.f) * v);
  float Av = 2.f - 2.f * __cosf((4.f * PIF / 1024.f) * v);
  float Bu = 2.f + 2.f * __cosf((2.f * PIF / 1024.f) * u);
  float grad = Au * Bv * Bv + Av * Bu * Bu;

  cpx f = FF[idx], l = FL[idx];
  float denom = f.x * f.x + f.y * f.y + 5.f + 5.f * grad;
  float invd = 1.f / denom;
  float nr = f.x * l.x + f.y * l.y;    // conj(FF)*FL
  float ni = f.x * l.y - f.y * l.x;
  Qc[idx] = cmk(nr * invd, -ni * invd);   // conj(Q)
}

// ---------------------------------------------------------------------------
// 5) psf = real(FFT2(conj Q))/N^2, fftshift, threshold.
// ---------------------------------------------------------------------------
__global__ __launch_bounds__(256) void finalize_kernel(const cpx* __restrict__ W,
                                                       float* __restrict__ p) {
  int idx = blockIdx.x * blockDim.x + threadIdx.x;
  if (idx >= SS) return;
  int r = idx >> 10, c = idx & (SDIM - 1);
  int rr = (r + 512) & 1023, cc = (c + 512) & 1023;
  float v = W[(size_t)rr * SDIM + cc].x * (1.f / 1048576.f);
  p[idx] = (v < 1e-4f) ? 0.f : v;
}

// ---------------------------------------------------------------------------
// 6) Flat argmax with JAX first-index tie-break (contiguous per-thread chunks).
// ---------------------------------------------------------------------------
__global__ __launch_bounds__(1024) void argmax_kernel(const float* __restrict__ p,
                                                      int* __restrict__ outrc) {
  __shared__ float sv[1024];
  __shared__ int   sx[1024];
  int t = threadIdx.x;
  float best = -INFINITY; int bidx = t * 1024;
  for (int i = 0; i < 1024; i++) {
    float v = p[t * 1024 + i];
    if (v > best) { best = v; bidx = t * 1024 + i; }
  }
  sv[t] = best; sx[t] = bidx;
  __syncthreads();
  for (int o = 512; o > 0; o >>= 1) {
    if (t < o) {
      if (sv[t + o] > sv[t] || (sv[t + o] == sv[t] && sx[t + o] < sx[t])) {
        sv[t] = sv[t + o]; sx[t] = sx[t + o];
      }
    }
    __syncthreads();
  }
  if (t == 0) { outrc[0] = sx[0] >> 10; outrc[1] = sx[0] & 1023; }
}

// ---------------------------------------------------------------------------
// 7) 41x41 crop (dynamic_slice start-clamp) + normalize by patch sum.
// ---------------------------------------------------------------------------
__global__ __launch_bounds__(512) void crop_kernel(const float* __restrict__ p,
                                                   const int* __restrict__ rc,
                                                   float* __restrict__ out) {
  __shared__ float sd[512];
  int t = threadIdx.x;
  int r0 = rc[0] - 20, c0 = rc[1] - 20;
  r0 = r0 < 0 ? 0 : (r0 > SDIM - 41 ? SDIM - 41 : r0);
  c0 = c0 < 0 ? 0 : (c0 > SDIM - 41 ? SDIM - 41 : c0);
  float local = 0.f;
  for (int i = t; i < 1681; i += 512)
    local += p[(size_t)(r0 + i / 41) * SDIM + c0 + i % 41];
  sd[t] = local;
  __syncthreads();
  for (int o = 256; o > 0; o >>= 1) {
    if (t < o) sd[t] += sd[t + o];
    __syncthreads();
  }
  float inv = 1.f / sd[0];
  for (int i = t; i < 1681; i += 512)
    out[i] = p[(size_t)(r0 + i / 41) * SDIM + c0 + i % 41] * inv;
}

// ---------------------------------------------------------------------------
extern "C" void kernel_launch(void* const* d_in, const int* in_sizes, int n_in,
                              void* d_out, int out_size, void* d_ws, size_t ws_size,
                              hipStream_t stream) {
  const float* lms  = (const float*)d_in[0];
  const float* fuse = (const float*)d_in[1];
  float* out = (float*)d_out;
  char* ws = (char*)d_ws;

  const size_t CB = (size_t)SS * sizeof(cpx);        // 8 MB complex buffer
  cpx*   A  = (cpx*)(ws);
  cpx*   Bb = (cpx*)(ws + CB);
  cpx*   Tt = (cpx*)(ws + 2 * CB);
  float* ra = (float*)(ws + 3 * CB);
  int*   rc = (int*)(ws + 3 * CB + (size_t)SS * sizeof(float));

  dim3 tb(32, 8), tg(32, 32);

  for (int c = 0; c < NCH; c++) {
    const float* lc = lms  + (size_t)c * SS;
    const float* fc = fuse + (size_t)c * SS;

    // edge blur both images -> complex A (lms), B (fuse)
    blur_kernel<<<SS / 256, 256, 0, stream>>>(lc, fc, A, Bb);

    // forward 2D FFTs: row -> transpose -> row  => transposed spectra
    fft_row_kernel<<<2048, 256, 0, stream>>>(A, Bb, 1024);   // rows of both
    transpose_kernel<<<tg, tb, 0, stream>>>(A, Tt);
    fft_row_kernel<<<1024, 256, 0, stream>>>(Tt, Tt, 1024);  // FL^T in Tt
    transpose_kernel<<<tg, tb, 0, stream>>>(Bb, A);
    fft_row_kernel<<<1024, 256, 0, stream>>>(A, A, 1024);    // FF^T in A

    // conj(Q)^T -> B
    wiener_kernel<<<SS / 256, 256, 0, stream>>>(Tt, A, Bb);

    // inverse via conj trick: FFT2(conj Q), natural orientation in A
    fft_row_kernel<<<1024, 256, 0, stream>>>(Bb, Bb, 1024);
    transpose_kernel<<<tg, tb, 0, stream>>>(Bb, A);
    fft_row_kernel<<<1024, 256, 0, stream>>>(A, A, 1024);

    finalize_kernel<<<SS / 256, 256, 0, stream>>>(A, ra);
    argmax_kernel<<<1, 1024, 0, stream>>>(ra, rc);
    crop_kernel<<<1, 512, 0, stream>>>(ra, rc, out + (size_t)c * 1681);
  }
}